// kernel_generated_0_62904091017393
// MI455X (gfx1250) — compile-verified
//
#include <hip/hip_runtime.h>

typedef float v2f __attribute__((ext_vector_type(2)));
typedef float v8f __attribute__((ext_vector_type(8)));

// Tile configuration
#define HT   14              // output rows per workgroup (112/14 = 8 chunks)
#define NR   (HT + 6)        // staged rows incl. 3+3 halo = 20
#define NW   8               // waves per workgroup (256 threads, wave32)

// LDS x tile: [rr][w][j]  (j innermost -> ds_load_b64 fetches j, j+1)
__global__ __launch_bounds__(256) void fused_shift_conv_wmma(
    const float* __restrict__ x,   // (128, 24, 112, 112)
    const float* __restrict__ w1,  // (12, 7, 9)
    const float* __restrict__ w2,  // (12, 9)
    float* __restrict__ out)       // (128, 24, 112, 112)
{
    __shared__ float ldsX[NR * 112 * 12];   // 26880 floats = 105 KB
    __shared__ float Wlds[84 * 16];         // [kj][i16], i>=12 zero-padded

    const int tid   = threadIdx.x;
    const int blk   = blockIdx.x;
    const int chunk = blk & 7;          // 8 h-chunks
    const int cIdx  = (blk >> 3) & 1;   // group c in [0,2)
    const int bIdx  = blk >> 4;         // batch
    const int h0    = chunk * HT;

    // ---- Build fused weight W[i][k][j] = sum_m w1[i,k,m]*w2[j,m] in LDS ----
    // Layout Wlds[kj*16 + i], kj = k*12 + j, padded i in [12,16) -> 0
    for (int t = tid; t < 84 * 16; t += 256) {
        const int i  = t & 15;
        const int kj = t >> 4;
        float v = 0.0f;
        if (i < 12) {
            const int k = kj / 12;
            const int j = kj - 12 * k;
            #pragma unroll
            for (int m = 0; m < 9; ++m)
                v += w1[i * 63 + k * 9 + m] * w2[j * 9 + m];
        }
        Wlds[t] = v;
    }

    // ---- Stage x tile into LDS with roll(+1,h) and zero halo resolved ----
    // rows: row = j*NR + rr ; each of 224 threads owns one w column, 2-way split
    {
        const int w = tid % 112;
        const int q = tid / 112;            // 0,1 active; q==2 idles
        if (q < 2) {
            const long xbase = ((long)bIdx * 24 + cIdx * 12) * 12544;
            for (int row = q; row < 12 * NR; row += 2) {
                const int j  = row / NR;
                const int rr = row - j * NR;
                const int hp = h0 - 3 + rr;        // h' in padded/rolled space
                float v = 0.0f;
                if (hp >= 0 && hp < 112) {
                    int srcrow = hp - 1;           // roll(+1) along h
                    if (srcrow < 0) srcrow += 112;
                    v = x[xbase + (long)j * 12544 + srcrow * 112 + w];
                }
                ldsX[(rr * 112 + w) * 12 + j] = v;
            }
        }
    }
    __syncthreads();

    // ---- WMMA compute: M=16 (12 channels), N=16 w-positions, K=84 ----
    const int wave = tid >> 5;
    const int lane = tid & 31;
    const int half = lane >> 4;     // 0: lanes 0-15, 1: lanes 16-31
    const int n    = lane & 15;     // N index (and M index for A loads)
    const int koff = half * 2;      // K offset within a 4-step per ISA layout

    // Preload all 21 A fragments (loop-invariant)
    v2f A[21];
    #pragma unroll
    for (int g = 0; g < 21; ++g) {
        A[g].x = Wlds[(4 * g + koff) * 16 + n];
        A[g].y = Wlds[(4 * g + koff + 1) * 16 + n];
    }

    const long obase0 = ((long)bIdx * 24 + cIdx * 12) * 12544;

    for (int tile = wave; tile < HT * 7; tile += NW) {
        const int hL = tile / 7;
        const int wt = tile - hL * 7;
        const int w0 = wt * 16;
        const int h  = h0 + hL;

        int ws = w0 + n - 1;                 // roll(+1) along w
        if (ws < 0) ws = 111;

        v8f c = {};
        #pragma unroll
        for (int g = 0; g < 21; ++g) {
            const int k  = g / 3;            // tap index (const-folded)
            const int jb = (g - k * 3) * 4;  // j base of this K-group
            const int j  = jb + koff;        // even -> 8B aligned b64 load
            const int rr = hL + k;
            const v2f bfrag = *(const v2f*)&ldsX[(rr * 112 + ws) * 12 + j];
            c = __builtin_amdgcn_wmma_f32_16x16x4_f32(
                    false, A[g], false, bfrag, (short)0, c, false, false);
        }

        // Store: lane covers w = w0+n; VGPR r covers channel i = r + half*8
        const long ob = obase0 + (long)h * 112 + w0 + n;
        #pragma unroll
        for (int r = 0; r < 8; ++r) {
            const int i = r + half * 8;
            if (i < 12) out[ob + (long)i * 12544] = c[r];
        }
    }
}

extern "C" void kernel_launch(void* const* d_in, const int* in_sizes, int n_in,
                              void* d_out, int out_size, void* d_ws, size_t ws_size,
                              hipStream_t stream) {
    (void)in_sizes; (void)n_in; (void)d_ws; (void)ws_size; (void)out_size;
    const float* x  = (const float*)d_in[0];
    const float* w1 = (const float*)d_in[1];
    const float* w2 = (const float*)d_in[2];
    float* out = (float*)d_out;

    // grid = 128 batches * 2 groups * 8 h-chunks
    fused_shift_conv_wmma<<<dim3(128 * 2 * 8), dim3(256), 0, stream>>>(x, w1, w2, out);
}